// AVODWH_CENTER_IN_31499290148938
// MI455X (gfx1250) — compile-verified
//
#include <hip/hip_runtime.h>
#include <hip/hip_bf16.h>
#include <math.h>

typedef __attribute__((ext_vector_type(2))) float v2f;
typedef __attribute__((ext_vector_type(8))) float v8f;

#define NIMG 8
#define NCLS 15
#define HH 128
#define WW 128
#define HW (HH * WW)
#define HWC (HW * NCLS)
#define KTOP 400
#define KOUT 100
#define PRE_T 0.05f
#define NMS_T 0.1f

// ---------------------------------------------------------------- kernel A:
// flat scores: p = sigmoid(cls), s = p*sigmoid(conf); flat = p>thr ? s : -1
__global__ void score_kernel(const float* __restrict__ box_cls,
                             const float* __restrict__ confs,
                             float* __restrict__ flat) {
    int t = blockIdx.x * blockDim.x + threadIdx.x;
    const int total = NIMG * HWC;
    if (t >= total) return;
    int n   = t / HWC;
    int rem = t - n * HWC;
    int li  = rem / NCLS;
    int ci  = rem - li * NCLS;
    float logit = box_cls[(size_t)n * NCLS * HW + (size_t)ci * HW + li];
    float p  = 1.0f / (1.0f + expf(-logit));
    float cf = confs[(size_t)n * HW + li];
    float s  = p * (1.0f / (1.0f + expf(-cf)));
    flat[t]  = (p > PRE_T) ? s : -1.0f;
}

// ---------------------------------------------------------------- kernel B:
// per-image iterative top-400 (L2-resident slice; LDS argmax reduction)
__global__ void topk_kernel(float* __restrict__ flat,
                            int* __restrict__ selIdx,
                            float* __restrict__ selVal) {
    __shared__ float shv[256];
    __shared__ int   shi[256];
    int n = blockIdx.x, tid = threadIdx.x;
    float* base = flat + (size_t)n * HWC;
    for (int r = 0; r < KTOP; ++r) {
        float lv = -1e30f; int li_ = 0x7fffffff;
        for (int k = tid; k < HWC; k += 256) {
            __builtin_prefetch(base + k + 4096, 0, 0);  // global_prefetch_b8
            float v = base[k];
            if (v > lv || (v == lv && k < li_)) { lv = v; li_ = k; }
        }
        shv[tid] = lv; shi[tid] = li_;
        __syncthreads();
        for (int s = 128; s > 0; s >>= 1) {
            if (tid < s) {
                float ov = shv[tid + s]; int oi = shi[tid + s];
                if (ov > shv[tid] || (ov == shv[tid] && oi < shi[tid])) {
                    shv[tid] = ov; shi[tid] = oi;
                }
            }
            __syncthreads();
        }
        if (tid == 0) {
            int bi = shi[0];
            selIdx[n * KTOP + r] = bi;
            selVal[n * KTOP + r] = shv[0];
            base[bi] = -2.0f;   // remove winner
        }
        __threadfence_block();
        __syncthreads();
    }
}

// ---------------------------------------------------------------- kernel C:
// decode selected detections into rotated boxes (cx,cy,w,h,ang)
__global__ void decode_kernel(const float* __restrict__ locations,
                              const float* __restrict__ box_reg,
                              const float* __restrict__ center,
                              const int* __restrict__ selIdx,
                              float* __restrict__ selVal,
                              float* __restrict__ boxes,
                              int* __restrict__ labels,
                              int* __restrict__ validArr) {
    int t = blockIdx.x * blockDim.x + threadIdx.x;
    if (t >= NIMG * KTOP) return;
    int n = t / KTOP;
    int idx = selIdx[t];
    float v = selVal[t];
    int li = idx / NCLS, ci = idx - li * NCLS;
    bool valid = v > 0.0f;
    const float* reg = box_reg + (size_t)n * 4 * HW;
    const float* ctr = center + (size_t)n * 2 * HW;
    float r0 = reg[li], r1 = reg[HW + li], r2 = reg[2 * HW + li], r3 = reg[3 * HW + li];
    float ce0 = ctr[li], ce1 = ctr[HW + li];
    float lo0 = locations[2 * li], lo1 = locations[2 * li + 1];
    float w = r0 + r1, h = r2 + r3;
    float cx = lo0 + ce0, cy = lo1 + ce1;
    float x1 = cx - 0.5f * w, y1 = cy - 0.5f * h;
    float x2 = cx + 0.5f * w, y2 = cy + 0.5f * h;
    float pw = r0, ph = r2;
    float X[4] = {x1 + pw, x2, x2 - pw, x1};
    float Y[4] = {y1, y1 + ph, y2, y2 - ph};
    float ang = atan2f(-(X[1] - X[0]), Y[1] - Y[0]);
    float mx = 0.25f * (X[0] + X[1] + X[2] + X[3]);
    float my = 0.25f * (Y[0] + Y[1] + Y[2] + Y[3]);
    float c = cosf(ang), s = sinf(ang);
    float xrmin = 1e30f, xrmax = -1e30f, yrmin = 1e30f, yrmax = -1e30f;
#pragma unroll
    for (int k = 0; k < 4; k++) {
        float dx = X[k] - mx, dy = Y[k] - my;
        float xr = c * dx + s * dy;
        float yr = -s * dx + c * dy;
        xrmin = fminf(xrmin, xr); xrmax = fmaxf(xrmax, xr);
        yrmin = fminf(yrmin, yr); yrmax = fmaxf(yrmax, yr);
    }
    float b0 = mx, b1 = my, b2 = xrmax - xrmin, b3 = yrmax - yrmin, b4 = ang;
    if (!valid) { b0 = b1 = b2 = b3 = b4 = 0.0f; }
    float* bo = boxes + (size_t)t * 5;
    bo[0] = b0; bo[1] = b1; bo[2] = b2; bo[3] = b3; bo[4] = b4;
    labels[t]   = ci;
    validArr[t] = valid ? 1 : 0;
    selVal[t]   = valid ? sqrtf(fmaxf(v, 0.0f)) : 0.0f;
}

// ------------------------------------------------------- rotated-box geometry
__device__ __forceinline__ void rot_corners(const float* b, float* X, float* Y) {
    const float dx[4] = {-0.5f, 0.5f, 0.5f, -0.5f};
    const float dy[4] = {-0.5f, -0.5f, 0.5f, 0.5f};
    float c = cosf(b[4]), s = sinf(b[4]);
#pragma unroll
    for (int k = 0; k < 4; k++) {
        float rx = dx[k] * b[2], ry = dy[k] * b[3];
        X[k] = b[0] + rx * c - ry * s;
        Y[k] = b[1] + rx * s + ry * c;
    }
}

__device__ float quad_inter_area(const float* bi, const float* bj) {
    float ax[4], ay[4], bx[4], by[4];
    rot_corners(bi, ax, ay);
    rot_corners(bj, bx, by);
    float px[12], py[12];
    int cnt = 4;
    for (int k = 0; k < 4; k++) { px[k] = ax[k]; py[k] = ay[k]; }
    for (int e = 0; e < 4; e++) {
        float x0 = bx[e], y0 = by[e];
        float x1 = bx[(e + 1) & 3], y1 = by[(e + 1) & 3];
        float ex = x1 - x0, ey = y1 - y0;
        float qx[12], qy[12];
        int m = 0;
        for (int i = 0; i < cnt; i++) {
            int j = (i + 1 == cnt) ? 0 : i + 1;
            float di = ex * (py[i] - y0) - ey * (px[i] - x0);
            float dj = ex * (py[j] - y0) - ey * (px[j] - x0);
            bool ii = di >= 0.0f, ij = dj >= 0.0f;
            if (ii) { qx[m] = px[i]; qy[m] = py[i]; m++; }
            if (ii != ij) {
                float t = di / (di - dj);
                qx[m] = px[i] + t * (px[j] - px[i]);
                qy[m] = py[i] + t * (py[j] - py[i]);
                m++;
            }
        }
        for (int i = 0; i < m; i++) { px[i] = qx[i]; py[i] = qy[i]; }
        cnt = m;
        if (cnt == 0) break;
    }
    if (cnt < 3) return 0.0f;
    float area = 0.0f;
    for (int i = 0; i < cnt; i++) {
        int j = (i + 1 == cnt) ? 0 : i + 1;
        area += px[i] * py[j] - px[j] * py[i];
    }
    return 0.5f * fabsf(area);
}

// ---------------------------------------------------------------- kernel D:
// 16x16 IoU tile per wave32. V_WMMA_F32_16X16X4_F32 computes the Gram-matrix
// of u_i.v_j = -2cx_i*cx_j + cy_i - 2|c_i|^2*cy_j + |c_j|^2, used as a
// d2 <= thr^2 prefilter before exact polygon clipping.
__global__ void __launch_bounds__(32)
iou_kernel(const float* __restrict__ boxes,
           const int* __restrict__ labels,
           const int* __restrict__ validArr,
           float* __restrict__ iou) {
    int lane = threadIdx.x;
    int tc = blockIdx.x, tr = blockIdx.y, n = blockIdx.z;
    __shared__ float RB[16][5], CB[16][5];
    __shared__ float RDg[16], CDg[16];
    __shared__ int RL[16], CL[16], RV[16], CV[16];
    int half = lane >> 4, l15 = lane & 15;
    {
        int g = (half == 0) ? (tr * 16 + l15) : (tc * 16 + l15);
        const float* b = boxes + ((size_t)n * KTOP + g) * 5;
        float b0 = b[0], b1 = b[1], b2 = b[2], b3 = b[3], b4 = b[4];
        int lb = labels[n * KTOP + g];
        int vd = validArr[n * KTOP + g];
        float dg = sqrtf(b2 * b2 + b3 * b3);
        if (half == 0) {
            RB[l15][0] = b0; RB[l15][1] = b1; RB[l15][2] = b2; RB[l15][3] = b3; RB[l15][4] = b4;
            RDg[l15] = dg; RL[l15] = lb; RV[l15] = vd;
        } else {
            CB[l15][0] = b0; CB[l15][1] = b1; CB[l15][2] = b2; CB[l15][3] = b3; CB[l15][4] = b4;
            CDg[l15] = dg; CL[l15] = lb; CV[l15] = vd;
        }
    }
    __syncthreads();

    // A (16x4): row m = l15 features (cx, cy, |c|^2, 1); lanes 0-15 hold K0,K1,
    // lanes 16-31 hold K2,K3.  B (4x16): col j = l15 features (-2cx,1,-2cy,|c|^2);
    // VGPR0 = K0 (lanes 0-15) / K2 (lanes 16-31), VGPR1 = K1 / K3.
    float acx = RB[l15][0], acy = RB[l15][1];
    float an2 = acx * acx + acy * acy;
    float bcx = CB[l15][0], bcy = CB[l15][1];
    float bn2 = bcx * bcx + bcy * bcy;
    v2f a, b;
    if (half == 0) {
        a.x = acx;          a.y = acy;
        b.x = -2.0f * bcx;  b.y = 1.0f;
    } else {
        a.x = an2;          a.y = 1.0f;
        b.x = -2.0f * bcy;  b.y = bn2;
    }
    v8f c = {};
    c = __builtin_amdgcn_wmma_f32_16x16x4_f32(false, a, false, b, (short)0, c,
                                              false, false);

#pragma unroll
    for (int v = 0; v < 8; ++v) {
        int m = half * 8 + v;            // C/D layout: VGPR v -> M=v (lo half), v+8 (hi)
        int I = tr * 16 + m, J = tc * 16 + l15;
        float d2 = c[v];
        float out = 0.0f;
        float thr = 0.5f * (RDg[m] + CDg[l15]);
        if (RV[m] && CV[l15] && RL[m] == CL[l15] && d2 <= thr * thr + 1e-3f) {
            float bi[5] = {RB[m][0], RB[m][1], RB[m][2], RB[m][3], RB[m][4]};
            float bj[5] = {CB[l15][0], CB[l15][1], CB[l15][2], CB[l15][3], CB[l15][4]};
            float inter = quad_inter_area(bi, bj);
            float un = bi[2] * bi[3] + bj[2] * bj[3] - inter;
            out = (un > 0.0f) ? inter / un : 0.0f;
        }
        iou[((size_t)n * KTOP + I) * KTOP + J] = out;
    }
}

// ---------------------------------------------------------------- kernel E:
// greedy NMS (serial by nature) + top-100 + output packing; one WG per image
__global__ void nms_kernel(const float* __restrict__ selVal,
                           const float* __restrict__ boxes,
                           const int* __restrict__ labels,
                           const int* __restrict__ validArr,
                           const float* __restrict__ iou,
                           float* __restrict__ out) {
    int n = blockIdx.x, tid = threadIdx.x;
    __shared__ float S[KTOP];
    __shared__ int SUP[KTOP], KEEP[KTOP];
    __shared__ float shv[256];
    __shared__ int shi[256];
    __shared__ float bV;
    __shared__ int bI, hasS;

    for (int k = tid; k < KTOP; k += 256) {
        S[k] = selVal[n * KTOP + k];
        SUP[k] = validArr[n * KTOP + k] ? 0 : 1;
        KEEP[k] = 0;
    }
    __syncthreads();

    for (int it = 0; it < KTOP; ++it) {
        float lv = -1e30f; int li_ = KTOP;
        for (int k = tid; k < KTOP; k += 256) {
            float v = SUP[k] ? -1e30f : S[k];
            if (v > lv || (v == lv && k < li_)) { lv = v; li_ = k; }
        }
        shv[tid] = lv; shi[tid] = li_;
        __syncthreads();
        for (int s = 128; s > 0; s >>= 1) {
            if (tid < s) {
                float ov = shv[tid + s]; int oi = shi[tid + s];
                if (ov > shv[tid] || (ov == shv[tid] && oi < shi[tid])) {
                    shv[tid] = ov; shi[tid] = oi;
                }
            }
            __syncthreads();
        }
        if (tid == 0) {
            bV = shv[0]; bI = shi[0];
            hasS = (bV > -1e29f && bI < KTOP) ? 1 : 0;
        }
        __syncthreads();
        if (!hasS) break;
        const float* row = iou + ((size_t)n * KTOP + bI) * KTOP;
        for (int k = tid; k < KTOP; k += 256)
            if (row[k] > NMS_T) SUP[k] = 1;
        if (tid == 0) { KEEP[bI] = 1; SUP[bI] = 1; }
        __syncthreads();
    }

    for (int k = tid; k < KTOP; k += 256) S[k] = KEEP[k] ? S[k] : -1.0f;
    __syncthreads();

    const int boxesOff  = 0;
    const int scoresOff = NIMG * KOUT * 5;
    const int labelsOff = scoresOff + NIMG * KOUT;
    const int vmaskOff  = labelsOff + NIMG * KOUT;
    for (int r = 0; r < KOUT; ++r) {
        float lv = -1e30f; int li_ = KTOP;
        for (int k = tid; k < KTOP; k += 256) {
            float v = S[k];
            if (v > lv || (v == lv && k < li_)) { lv = v; li_ = k; }
        }
        shv[tid] = lv; shi[tid] = li_;
        __syncthreads();
        for (int s = 128; s > 0; s >>= 1) {
            if (tid < s) {
                float ov = shv[tid + s]; int oi = shi[tid + s];
                if (ov > shv[tid] || (ov == shv[tid] && oi < shi[tid])) {
                    shv[tid] = ov; shi[tid] = oi;
                }
            }
            __syncthreads();
        }
        if (tid == 0) {
            float v = shv[0]; int j = shi[0];
            bool m = v > 0.0f;
            const float* bo = boxes + ((size_t)n * KTOP + j) * 5;
#pragma unroll
            for (int q = 0; q < 5; q++)
                out[boxesOff + ((size_t)n * KOUT + r) * 5 + q] = m ? bo[q] : 0.0f;
            out[scoresOff + n * KOUT + r] = m ? v : 0.0f;
            out[labelsOff + n * KOUT + r] = (float)labels[n * KTOP + j];
            out[vmaskOff + n * KOUT + r] = m ? 1.0f : 0.0f;
            S[j] = -2.0f;
        }
        __syncthreads();
    }
}

extern "C" void kernel_launch(void* const* d_in, const int* in_sizes, int n_in,
                              void* d_out, int out_size, void* d_ws, size_t ws_size,
                              hipStream_t stream) {
    (void)in_sizes; (void)n_in; (void)out_size; (void)ws_size;
    const float* locations = (const float*)d_in[0];
    const float* box_cls   = (const float*)d_in[1];
    const float* box_reg   = (const float*)d_in[2];
    const float* center    = (const float*)d_in[3];
    const float* confs     = (const float*)d_in[4];
    float* out = (float*)d_out;

    float* ws = (float*)d_ws;
    size_t off = 0;
    float* flat   = ws + off; off += (size_t)NIMG * HWC;        // 1,966,080
    int*   selIdx = (int*)(ws + off); off += NIMG * KTOP;       // 3,200
    float* selVal = ws + off; off += NIMG * KTOP;               // 3,200
    float* boxes  = ws + off; off += NIMG * KTOP * 5;           // 16,000
    int*   labels = (int*)(ws + off); off += NIMG * KTOP;       // 3,200
    int*   validA = (int*)(ws + off); off += NIMG * KTOP;       // 3,200
    float* iou    = ws + off; off += (size_t)NIMG * KTOP * KTOP; // 1,280,000

    const int totalA = NIMG * HWC;
    score_kernel<<<(totalA + 255) / 256, 256, 0, stream>>>(box_cls, confs, flat);
    topk_kernel<<<NIMG, 256, 0, stream>>>(flat, selIdx, selVal);
    decode_kernel<<<(NIMG * KTOP + 255) / 256, 256, 0, stream>>>(
        locations, box_reg, center, selIdx, selVal, boxes, labels, validA);
    dim3 gD(KTOP / 16, KTOP / 16, NIMG);
    iou_kernel<<<gD, 32, 0, stream>>>(boxes, labels, validA, iou);
    nms_kernel<<<NIMG, 256, 0, stream>>>(selVal, boxes, labels, validA, iou, out);
}